// PtModule_76166950027759
// MI455X (gfx1250) — compile-verified
//
#include <hip/hip_runtime.h>

// Elementwise y = (x+1)^2 over 8192x8192 fp32.
// Memory-bound: 512 MB traffic -> ~22 us floor at 23.3 TB/s. WMMA/TDM
// inapplicable (no contraction; LDS staging adds latency for use-once data).
//
// Round-1 fix: remove per-item bounds checks (they generated exec-mask
// branches + 32 VGPR zero-init movs) by launching an exact, guard-free grid
// for the divisible bulk and a separate guarded tail kernel for leftovers.
// Constant intra-block offsets fold into the VMEM immediate (offset:k*4096),
// so each thread computes one address and issues 8 NT B128 loads + 8 NT B128
// stores back-to-back.

typedef float v4f __attribute__((ext_vector_type(4)));

constexpr int BLOCK = 256;   // 8 wave32s per workgroup
constexpr int ITEMS = 8;     // 8 x float4 = 32 floats per thread (32 KB/block)

__global__ __launch_bounds__(BLOCK) void ew_sq_bulk(
    const v4f* __restrict__ x, v4f* __restrict__ y) {
  const long base = (long)blockIdx.x * (BLOCK * ITEMS) + threadIdx.x;
  const v4f* __restrict__ xp = x + base;
  v4f* __restrict__       yp = y + base;

  v4f v[ITEMS];
  // 8 outstanding global_load_b128 th:NT from one base address; offsets are
  // compile-time constants (k*4096 bytes) -> VMEM immediate offsets.
#pragma unroll
  for (int k = 0; k < ITEMS; ++k) {
    v[k] = __builtin_nontemporal_load(&xp[k * BLOCK]);
  }
#pragma unroll
  for (int k = 0; k < ITEMS; ++k) {
    v4f t = v[k] + 1.0f;
    __builtin_nontemporal_store(t * t, &yp[k * BLOCK]);
  }
}

// Guarded scalar tail for any elements past the divisible bulk (zero blocks
// for the 8192x8192 case; kept for generality/correctness).
__global__ __launch_bounds__(BLOCK) void ew_sq_tail(
    const float* __restrict__ x, float* __restrict__ y, long start, long n) {
  const long i = start + (long)blockIdx.x * BLOCK + threadIdx.x;
  if (i < n) {
    float t = x[i] + 1.0f;
    __builtin_nontemporal_store(t * t, &y[i]);
  }
}

extern "C" void kernel_launch(void* const* d_in, const int* in_sizes, int n_in,
                              void* d_out, int out_size, void* d_ws, size_t ws_size,
                              hipStream_t stream) {
  (void)n_in; (void)d_ws; (void)ws_size; (void)out_size;
  const float* x = (const float*)d_in[0];
  float*       y = (float*)d_out;

  const long n  = (long)in_sizes[0];          // 67,108,864
  const long n4 = n >> 2;                     // float4 chunks in the vector bulk
  const long per_block   = (long)BLOCK * ITEMS;     // 2048 float4 per block
  const long full_blocks = n4 / per_block;          // 8192 (exact for this shape)
  const long done        = full_blocks * per_block * 4;  // elements covered

  if (full_blocks > 0) {
    ew_sq_bulk<<<(int)full_blocks, BLOCK, 0, stream>>>(
        (const v4f*)x, (v4f*)y);
  }
  const long rem = n - done;                  // 0 for 8192x8192
  if (rem > 0) {
    const int tail_blocks = (int)((rem + BLOCK - 1) / BLOCK);
    ew_sq_tail<<<tail_blocks, BLOCK, 0, stream>>>(x, y, done, n);
  }
}